// DilateAttention_26491358282246
// MI455X (gfx1250) — compile-verified
//
#include <hip/hip_runtime.h>

typedef __attribute__((ext_vector_type(2))) float v2f;
typedef __attribute__((ext_vector_type(8))) float v8f;

#define HH 56
#define WW 56
#define PP (HH * WW)
#define DD 768
#define HD 64
#define NH 12
#define SCALE 0.125f   // 64^-0.5

// One wave (32 threads) computes a tile of 16 pixels (one image row segment)
// for one (batch, head). QK^T runs on the matrix pipe as fp32 WMMA
// (V_WMMA_F32_16X16X4_F32), softmax + AV on VALU with coalesced streaming.
__launch_bounds__(32)
__global__ void DilateAttention_26491358282246_kernel(
    const float* __restrict__ q,
    const float* __restrict__ k,
    const float* __restrict__ v,
    float* __restrict__ out)
{
    __shared__ float S_lds[3][16][20];  // 3 bands x 16 pixels x 18(+pad) cols
    __shared__ float w_lds[16][9];      // softmax weights per pixel

    const int lane = threadIdx.x;       // 0..31
    const int tx   = blockIdx.x;        // 0..3  (x tile)
    const int y    = blockIdx.y;        // 0..55 (image row)
    const int bh   = blockIdx.z;        // 0..95
    const int b    = bh / NH;
    const int head = bh % NH;
    const int x0   = tx * 16;

    const size_t chanBase = ((size_t)b * DD + (size_t)head * HD) * PP;
    const float* qg = q + chanBase;
    const float* kg = k + chanBase;
    const float* vg = v + chanBase;

    const int half = lane >> 4;         // 0 or 1
    const int l16  = lane & 15;
    const int kh   = half * 2;          // K sub-offset inside each K=4 step

    // ---------------- A fragments: Q tile [16 pixels x 64 ch] ----------------
    const int  m        = l16;                    // pixel-in-tile (as A row)
    const bool pixValid = (x0 + m) < WW;
    const int  flatQ    = y * WW + x0 + m;
    const float* qp     = qg + (pixValid ? flatQ : 0);   // clamped, never OOB
    v2f aF[16];
#pragma unroll
    for (int t = 0; t < 16; ++t) {
        aF[t].x = qp[(size_t)(4 * t + kh) * PP];
        aF[t].y = qp[(size_t)(4 * t + kh + 1) * PP];
    }

    // Second N-tile (cols 16..17) only needed if pixel m>=14 is in-image.
    const bool needC1 = (x0 + 14) < WW;

    // ---------------- QK^T: 3 column bands, fp32 WMMA ----------------
#pragma unroll
    for (int band = 0; band < 3; ++band) {
        const int  yy       = y + band - 1;
        const bool rowValid = (yy >= 0) && (yy < HH);
        const int  nC       = needC1 ? 2 : 1;
        for (int c = 0; c < nC; ++c) {
            const int  n        = c * 16 + l16;        // band column (as B col)
            const int  xx       = x0 + n - 1;
            const bool colValid = rowValid && (xx >= 0) && (xx < WW);
            const int  flatK    = yy * WW + xx;
            const float* kp     = kg + (colValid ? flatK : 0);
            v8f acc = {};
#pragma unroll
            for (int t = 0; t < 16; ++t) {
                float b0 = kp[(size_t)(4 * t + kh) * PP];
                float b1 = kp[(size_t)(4 * t + kh + 1) * PP];
                v2f bF;
                bF.x = colValid ? b0 : 0.0f;   // zero-fill => padded logits = 0
                bF.y = colValid ? b1 : 0.0f;
                acc = __builtin_amdgcn_wmma_f32_16x16x4_f32(
                        false, aF[t], false, bF, (short)0, acc, false, false);
            }
            // C layout: vgpr r -> M = r + 8*half, N = l16 (+16c). Spill band.
#pragma unroll
            for (int r = 0; r < 8; ++r) {
                const int mm = r + half * 8;
                const int nn = c * 16 + l16;
                if (nn < 18) S_lds[band][mm][nn] = acc[r];
            }
        }
    }
    __syncthreads();

    // ---------------- softmax over 9 taps (lanes 0..15) ----------------
    if (lane < 16 && pixValid) {
        float l[9];
#pragma unroll
        for (int i = 0; i < 3; ++i)
#pragma unroll
            for (int j = 0; j < 3; ++j)
                l[i * 3 + j] = S_lds[i][m][m + j] * SCALE;
        float mx = l[0];
#pragma unroll
        for (int t = 1; t < 9; ++t) mx = fmaxf(mx, l[t]);
        float s = 0.0f;
#pragma unroll
        for (int t = 0; t < 9; ++t) { l[t] = __expf(l[t] - mx); s += l[t]; }
        const float inv = 1.0f / s;
#pragma unroll
        for (int t = 0; t < 9; ++t) w_lds[m][t] = l[t] * inv;
    }
    __syncthreads();

    // ---------------- AV: all 32 lanes, 32 channels each ----------------
    float        wgt[9];
    const float* vp[9];
#pragma unroll
    for (int i = 0; i < 3; ++i) {
#pragma unroll
        for (int j = 0; j < 3; ++j) {
            const int  tap = i * 3 + j;
            const int  yy  = y + i - 1;
            const int  xx  = x0 + m + j - 1;
            const bool tv  = pixValid && yy >= 0 && yy < HH && xx >= 0 && xx < WW;
            vp[tap]  = vg + (tv ? (yy * WW + xx) : 0);
            wgt[tap] = tv ? w_lds[m][tap] : 0.0f;   // padded taps see v = 0
        }
    }
    const int hbase = half * 32;
    float acc[32];
#pragma unroll
    for (int h = 0; h < 32; ++h) acc[h] = 0.0f;
#pragma unroll
    for (int tap = 0; tap < 9; ++tap) {
        const float* vt = vp[tap] + (size_t)hbase * PP;
        const float  ww = wgt[tap];
#pragma unroll
        for (int h = 0; h < 32; ++h)
            acc[h] = fmaf(ww, vt[(size_t)h * PP], acc[h]);
    }

    // out[b, y, x, head*64 + ch]: contiguous per pixel -> b128 stores
    if (pixValid) {
        float* op = out + (((size_t)b * HH + y) * WW + (x0 + m)) * DD
                        + head * HD + hbase;
#pragma unroll
        for (int h4 = 0; h4 < 8; ++h4) {
            float4 val = make_float4(acc[4 * h4 + 0], acc[4 * h4 + 1],
                                     acc[4 * h4 + 2], acc[4 * h4 + 3]);
            *reinterpret_cast<float4*>(op + 4 * h4) = val;
        }
    }
}

extern "C" void kernel_launch(void* const* d_in, const int* in_sizes, int n_in,
                              void* d_out, int out_size, void* d_ws, size_t ws_size,
                              hipStream_t stream) {
    const float* q = (const float*)d_in[0];
    const float* k = (const float*)d_in[1];
    const float* v = (const float*)d_in[2];
    float* out = (float*)d_out;

    dim3 grid(4, HH, 8 * NH);   // x tiles, image rows, B*heads
    dim3 block(32, 1, 1);       // one wave32 per tile
    DilateAttention_26491358282246_kernel<<<grid, block, 0, stream>>>(q, k, v, out);
}